// QPSegmentChargeNormalization_76063870812744
// MI455X (gfx1250) — compile-verified
//
#include <hip/hip_runtime.h>
#include <hip/hip_bf16.h>

// CDNA5 / gfx1250 wave32. One 256-thread block (8 waves) per batch row.
// Each wave owns a 16-atom tile; V_WMMA_F32_16X16X4_F32 with an all-ones B
// performs the F=16 row reduction (D = E * J replicates row sums across lanes).

typedef float v2f __attribute__((ext_vector_type(2)));
typedef float v8f __attribute__((ext_vector_type(8)));

#define NB 8192
#define NA 128
#define NF 16
#define NS 8

__global__ __launch_bounds__(256, 4)
void QPSegmentChargeNormalization_kernel(const float* __restrict__ c,
                                         const float* __restrict__ q,
                                         const int*   __restrict__ seg,
                                         const float* __restrict__ charges,
                                         float*       __restrict__ out)
{
    __shared__ int   s_seg[NA];
    __shared__ float s_qc[NS];
    __shared__ float s_qq[NS];
    __shared__ float s_lam[NS];

    const int b     = blockIdx.x;
    const int tid   = threadIdx.x;
    const int lane  = tid & 31;
    const int wave  = tid >> 5;             // 0..7 : 16-atom tile per wave
    const int m     = lane & 15;            // atom row within tile (A-matrix M)
    const int khalf = (lane >> 4) << 1;     // lanes 0-15 -> K {4j,4j+1}; 16-31 -> {4j+2,4j+3}

    if (tid < NA) s_seg[tid] = seg[(size_t)b * NA + tid];
    if (tid < NS) { s_qc[tid] = 0.0f; s_qq[tid] = 0.0f; }

    const int    atom    = wave * 16 + m;
    const size_t rowBase = ((size_t)b * NA + atom) * NF;

    // Each lane loads its 8 floats of c and q (float2 per K-group) — one tile
    // is covered exactly once per wave; registers are reused for the output pass.
    float2 cv[4], qv[4];
#pragma unroll
    for (int j = 0; j < 4; ++j) {
        const size_t off = rowBase + 4 * j + khalf;
        cv[j] = *(const float2*)(c + off);
        qv[j] = *(const float2*)(q + off);
    }

    __syncthreads();   // s_seg/s_qc/s_qq ready

    v2f ones; ones.x = 1.0f; ones.y = 1.0f;
    v8f acc_cq = {};
    v8f acc_qq = {};
#pragma unroll
    for (int j = 0; j < 4; ++j) {        // kb = 4j, accumulate K=16 in 4 steps
        v2f e; e.x = cv[j].x * qv[j].x; e.y = cv[j].y * qv[j].y;
        acc_cq = __builtin_amdgcn_wmma_f32_16x16x4_f32(
            false, e, false, ones, (short)0, acc_cq, false, false);
        v2f g; g.x = qv[j].x * qv[j].x; g.y = qv[j].y * qv[j].y;
        acc_qq = __builtin_amdgcn_wmma_f32_16x16x4_f32(
            false, g, false, ones, (short)0, acc_qq, false, false);
    }

    // D[m,n] is n-independent: lane n<16 holds rows 0..7 in VGPR 0..7,
    // lanes 16..31 hold rows 8..15. Use 4 lanes per wave for the LDS atomics.
    if (lane == 0 || lane == 16) {
        const int base = wave * 16 + ((lane >> 4) << 3);
#pragma unroll
        for (int r = 0; r < 8; ++r)
            atomicAdd(&s_qc[s_seg[base + r]], acc_cq[r]);
    }
    if (lane == 1 || lane == 17) {
        const int base = wave * 16 + ((lane >> 4) << 3);
#pragma unroll
        for (int r = 0; r < 8; ++r)
            atomicAdd(&s_qq[s_seg[base + r]], acc_qq[r]);
    }

    __syncthreads();

    if (tid < NS)
        s_lam[tid] = (s_qc[tid] - charges[(size_t)b * NS + tid]) / s_qq[tid];

    __syncthreads();

    const float lam = s_lam[s_seg[atom]];
    float* op = out + rowBase;
#pragma unroll
    for (int j = 0; j < 4; ++j) {
        float2 r;
        r.x = cv[j].x - qv[j].x * lam;
        r.y = cv[j].y - qv[j].y * lam;
        *(float2*)(op + 4 * j + khalf) = r;
    }
}

extern "C" void kernel_launch(void* const* d_in, const int* in_sizes, int n_in,
                              void* d_out, int out_size, void* d_ws, size_t ws_size,
                              hipStream_t stream) {
    const float* c       = (const float*)d_in[0];   // [B,A,F] f32
    const float* q       = (const float*)d_in[1];   // [B,A,F] f32
    const int*   seg     = (const int*)  d_in[2];   // [B,A]   int32
    const float* charges = (const float*)d_in[3];   // [B,S]   f32
    float*       out     = (float*)d_out;           // [B,A,F] f32
    (void)in_sizes; (void)n_in; (void)out_size; (void)d_ws; (void)ws_size;

    QPSegmentChargeNormalization_kernel<<<NB, 256, 0, stream>>>(c, q, seg, charges, out);
}